// Block_14087492730928
// MI455X (gfx1250) — compile-verified
//
#include <hip/hip_runtime.h>

// ---------------------------------------------------------------------------
// GNN message-passing layer for MI455X (gfx1250, wave32, WMMA).
//
// Working set (node table 51MB + inbox 51MB) is L2-resident (192MB), so the
// kernels are L2-gather / matrix-pipe / LDS bound, not HBM bound. fp32 is
// kept end-to-end via V_WMMA_F32_16X16X4_F32.
//
// Edge kernel: one wave = 32(M) x 128(N) tile (two 16-row A fragments share
// every B fragment -> DS traffic per WMMA halved). Node kernel: 16 x 128.
// K = 256 in 64 steps of 4.
//   A fragment (16x4 f32): lane(m = lane&15, h = lane>>4) holds
//       float2 = A[m][4*kk + 2h .. +1]   (one gathered global float2 load)
//   B fragment (4x16 f32): float2 = W[4*kk+2h][16*nt+n], W[4*kk+2h+1][...]
//       pre-packed in LDS in fragment order -> one ds_load_b64, conflict-free
//   C/D tile: acc[nt] v8f; element [v] of lane = D[v + 8h][16*nt + (lane&15)]
// ---------------------------------------------------------------------------

#define DIM 128     // IN_DIM == DIM_H
#define EPSV 1e-5f
#define WAVES 8     // waves per block

typedef __attribute__((ext_vector_type(2))) float v2f;
typedef __attribute__((ext_vector_type(8))) float v8f;

#define WMMA_F32(A, B, C) \
    __builtin_amdgcn_wmma_f32_16x16x4_f32(false, (A), false, (B), (short)0, (C), false, false)

// Pack row-major W[256][128] into LDS as per-lane WMMA B-fragments:
// float2 entry e = ((kk*8 + nt)*32 + lane):
//   .x = W[4*kk + 2*(lane>>4)    ][16*nt + (lane&15)]
//   .y = W[4*kk + 2*(lane>>4) + 1][16*nt + (lane&15)]
// 64 ksteps * 8 ntiles * 32 lanes * 2 floats = 32768 floats = 128 KB.
__device__ inline void pack_w_lds(const float* __restrict__ W, float* lds) {
    for (int e = threadIdx.x; e < 16384; e += blockDim.x) {
        int kk  = e >> 8;          // 0..63
        int rem = e & 255;
        int nt  = rem >> 5;        // 0..7
        int l   = rem & 31;        // 0..31
        int n   = nt * 16 + (l & 15);
        int k0  = kk * 4 + ((l >> 4) << 1);
        lds[2 * e]     = W[(size_t)k0 * DIM + n];
        lds[2 * e + 1] = W[(size_t)(k0 + 1) * DIM + n];
    }
}

__global__ void zero_f32(float* __restrict__ p, size_t n) {
    size_t i = (size_t)blockIdx.x * blockDim.x + threadIdx.x;
    size_t st = (size_t)gridDim.x * blockDim.x;
    for (; i < n; i += st) p[i] = 0.0f;
}

// LayerNorm one 16x128 accumulator tile in-register and scatter-add it into
// inbox rows given by the receiver index register (broadcast via shfl).
__device__ inline void ln_scatter_tile(v8f* acc, const float* bias,
                                       const float* gam, const float* bet,
                                       int ridx32, float* __restrict__ inboxB,
                                       int m, int h) {
#pragma unroll
    for (int v = 0; v < 8; ++v) {
        // row M = v + 8h; its 128 values live in element [v] of acc[0..7]
        // across the 16 lanes of this half.
        float x[8], s = 0.f, s2 = 0.f;
#pragma unroll
        for (int nt = 0; nt < 8; ++nt) {
            x[nt] = acc[nt][v] + bias[nt];
            s += x[nt]; s2 += x[nt] * x[nt];
        }
#pragma unroll
        for (int mask = 1; mask < 16; mask <<= 1) {
            s  += __shfl_xor(s,  mask, 32);
            s2 += __shfl_xor(s2, mask, 32);
        }
        float mu  = s * (1.0f / 128.0f);
        float var = s2 * (1.0f / 128.0f) - mu * mu;
        float inv = rsqrtf(var + EPSV);
        int dstRow = __shfl(ridx32, v + 8 * h, 32);  // receiver of row M
        float* __restrict__ dst = inboxB + (size_t)dstRow * DIM + m;
#pragma unroll
        for (int nt = 0; nt < 8; ++nt) {
            float y = (x[nt] - mu) * inv * gam[nt] + bet[nt];
            atomicAdd(dst + nt * 16, y);   // global_atomic_add_f32 (L2)
        }
    }
}

// -------------------------- edge / message kernel ---------------------------
// messages = LN(concat(nodes[s], nodes[r]) @ W_msg + b_msg) scattered
// (atomicAdd) into inbox[receiver]. 32 edges per wave.
__global__ __launch_bounds__(32 * WAVES) void edge_msg_kernel(
    const float* __restrict__ nodes, const long long* __restrict__ senders,
    const long long* __restrict__ receivers, const float* __restrict__ W_msg,
    const float* __restrict__ b_msg, const float* __restrict__ g1,
    const float* __restrict__ be1, float* __restrict__ inbox, int N, int E) {
    extern __shared__ float wlds[];
    pack_w_lds(W_msg, wlds);
    __syncthreads();

    const int b    = blockIdx.y;
    const int lane = threadIdx.x & 31;
    const int wid  = threadIdx.x >> 5;
    const int tile = blockIdx.x * WAVES + wid;   // 32-edge tiles
    if (tile >= (E >> 5)) return;   // wave-uniform: EXEC stays all-1 for WMMA

    const int m = lane & 15;
    const int h = lane >> 4;
    const int e0 = tile << 5;

    const size_t eb = (size_t)b * E + e0 + m;
    const long long sIdx0 = senders[eb];
    const long long rIdx0 = receivers[eb];
    const long long sIdx1 = senders[eb + 16];
    const long long rIdx1 = receivers[eb + 16];
    const size_t nb = (size_t)b * N;
    const float* __restrict__ s0 = nodes + (nb + (size_t)sIdx0) * DIM;
    const float* __restrict__ r0 = nodes + (nb + (size_t)rIdx0) * DIM;
    const float* __restrict__ s1 = nodes + (nb + (size_t)sIdx1) * DIM;
    const float* __restrict__ r1 = nodes + (nb + (size_t)rIdx1) * DIM;

    v8f acc0[8] = {};
    v8f acc1[8] = {};

    // K = 0..127: sender features
#pragma unroll 2
    for (int kk = 0; kk < 32; ++kk) {
        v2f a0 = *(const v2f*)(s0 + kk * 4 + 2 * h);
        v2f a1 = *(const v2f*)(s1 + kk * 4 + 2 * h);
        const float* wb = &wlds[(size_t)(kk * 8) * 64 + lane * 2];
#pragma unroll
        for (int nt = 0; nt < 8; ++nt) {
            v2f bf = *(const v2f*)(wb + nt * 64);
            acc0[nt] = WMMA_F32(a0, bf, acc0[nt]);
            acc1[nt] = WMMA_F32(a1, bf, acc1[nt]);
        }
    }
    // K = 128..255: receiver features
#pragma unroll 2
    for (int kk = 32; kk < 64; ++kk) {
        v2f a0 = *(const v2f*)(r0 + (kk - 32) * 4 + 2 * h);
        v2f a1 = *(const v2f*)(r1 + (kk - 32) * 4 + 2 * h);
        const float* wb = &wlds[(size_t)(kk * 8) * 64 + lane * 2];
#pragma unroll
        for (int nt = 0; nt < 8; ++nt) {
            v2f bf = *(const v2f*)(wb + nt * 64);
            acc0[nt] = WMMA_F32(a0, bf, acc0[nt]);
            acc1[nt] = WMMA_F32(a1, bf, acc1[nt]);
        }
    }

    float bias[8], gam[8], bet[8];
#pragma unroll
    for (int nt = 0; nt < 8; ++nt) {
        int n = nt * 16 + m;
        bias[nt] = b_msg[n]; gam[nt] = g1[n]; bet[nt] = be1[n];
    }

    float* inboxB = inbox + nb * DIM;
    ln_scatter_tile(acc0, bias, gam, bet, (int)rIdx0, inboxB, m, h);
    ln_scatter_tile(acc1, bias, gam, bet, (int)rIdx1, inboxB, m, h);
}

// --------------------------- node update kernel -----------------------------
// out = LN(concat(nodes, inbox) @ W_node + b_node). 16 nodes per wave
// (N % 32 != 0, and this kernel is only ~11% of FLOPs).
__global__ __launch_bounds__(32 * WAVES) void node_update_kernel(
    const float* __restrict__ nodes, const float* __restrict__ inbox,
    const float* __restrict__ W_node, const float* __restrict__ b_node,
    const float* __restrict__ g2, const float* __restrict__ be2,
    float* __restrict__ out, int N) {
    extern __shared__ float wlds[];
    pack_w_lds(W_node, wlds);
    __syncthreads();

    const int b    = blockIdx.y;
    const int lane = threadIdx.x & 31;
    const int wid  = threadIdx.x >> 5;
    const int tile = blockIdx.x * WAVES + wid;
    if (tile >= (N >> 4)) return;   // N % 16 == 0 (50000)

    const int m = lane & 15;
    const int h = lane >> 4;
    const int r0 = tile << 4;

    const float* __restrict__ srow = nodes + ((size_t)b * N + r0 + m) * DIM;
    const float* __restrict__ irow = inbox + ((size_t)b * N + r0 + m) * DIM;

    v8f acc[8] = {};
#pragma unroll 4
    for (int kk = 0; kk < 32; ++kk) {
        v2f a = *(const v2f*)(srow + kk * 4 + 2 * h);
        const float* wb = &wlds[(size_t)(kk * 8) * 64 + lane * 2];
#pragma unroll
        for (int nt = 0; nt < 8; ++nt) {
            v2f bf = *(const v2f*)(wb + nt * 64);
            acc[nt] = WMMA_F32(a, bf, acc[nt]);
        }
    }
#pragma unroll 4
    for (int kk = 32; kk < 64; ++kk) {
        v2f a = *(const v2f*)(irow + (kk - 32) * 4 + 2 * h);
        const float* wb = &wlds[(size_t)(kk * 8) * 64 + lane * 2];
#pragma unroll
        for (int nt = 0; nt < 8; ++nt) {
            v2f bf = *(const v2f*)(wb + nt * 64);
            acc[nt] = WMMA_F32(a, bf, acc[nt]);
        }
    }

    float bias[8], gam[8], bet[8];
#pragma unroll
    for (int nt = 0; nt < 8; ++nt) {
        int n = nt * 16 + m;
        bias[nt] = b_node[n]; gam[nt] = g2[n]; bet[nt] = be2[n];
    }

#pragma unroll
    for (int v = 0; v < 8; ++v) {
        float x[8], s = 0.f, s2 = 0.f;
#pragma unroll
        for (int nt = 0; nt < 8; ++nt) {
            x[nt] = acc[nt][v] + bias[nt];
            s += x[nt]; s2 += x[nt] * x[nt];
        }
#pragma unroll
        for (int mask = 1; mask < 16; mask <<= 1) {
            s  += __shfl_xor(s,  mask, 32);
            s2 += __shfl_xor(s2, mask, 32);
        }
        float mu  = s * (1.0f / 128.0f);
        float var = s2 * (1.0f / 128.0f) - mu * mu;
        float inv = rsqrtf(var + EPSV);
        float* __restrict__ dst =
            out + ((size_t)b * N + r0 + v + 8 * h) * DIM + m;
#pragma unroll
        for (int nt = 0; nt < 8; ++nt)
            dst[nt * 16] = (x[nt] - mu) * inv * gam[nt] + bet[nt];
    }
}

extern "C" void kernel_launch(void* const* d_in, const int* in_sizes, int n_in,
                              void* d_out, int out_size, void* d_ws, size_t ws_size,
                              hipStream_t stream) {
    const float*     nodes     = (const float*)d_in[0];
    const long long* senders   = (const long long*)d_in[1];  // int64
    const long long* receivers = (const long long*)d_in[2];  // int64
    const float*     W_msg     = (const float*)d_in[3];
    const float*     b_msg     = (const float*)d_in[4];
    const float*     W_node    = (const float*)d_in[5];
    const float*     b_node    = (const float*)d_in[6];
    const float*     g1        = (const float*)d_in[7];
    const float*     be1       = (const float*)d_in[8];
    const float*     g2        = (const float*)d_in[9];
    const float*     be2       = (const float*)d_in[10];
    float*           out       = (float*)d_out;

    const int B = 2;
    const int N = in_sizes[0] / (B * DIM);
    const int E = in_sizes[1] / B;

    float* inbox = (float*)d_ws;                 // B*N*DIM fp32 = 51.2 MB
    const size_t inboxN = (size_t)B * N * DIM;

    const size_t ldsBytes = 32768 * sizeof(float);   // 128 KB packed W
    hipFuncSetAttribute((const void*)edge_msg_kernel,
                        hipFuncAttributeMaxDynamicSharedMemorySize,
                        (int)ldsBytes);
    hipFuncSetAttribute((const void*)node_update_kernel,
                        hipFuncAttributeMaxDynamicSharedMemorySize,
                        (int)ldsBytes);

    // 1) zero the inbox accumulator (graph replays must not accumulate)
    zero_f32<<<1024, 256, 0, stream>>>(inbox, inboxN);

    // 2) edge messages: gather + GEMM + LN + scatter-add (32 edges / wave)
    dim3 gridE((E / 32 + WAVES - 1) / WAVES, B);
    edge_msg_kernel<<<gridE, 32 * WAVES, ldsBytes, stream>>>(
        nodes, senders, receivers, W_msg, b_msg, g1, be1, inbox, N, E);

    // 3) node update: GEMM + LN -> out (16 nodes / wave)
    dim3 gridN((N / 16 + WAVES - 1) / WAVES, B);
    node_update_kernel<<<gridN, 32 * WAVES, ldsBytes, stream>>>(
        nodes, inbox, W_node, b_node, g2, be2, out, N);
}